// MixtureOfExperts_v2_10703058502307
// MI455X (gfx1250) — compile-verified
//
#include <hip/hip_runtime.h>

#define INPUT_DIM    2304
#define SUBSPACE_DIM 64
#define ATOMS        512
#define NUM_EXPERTS  256
#define TOPK         4
// 2304 = 48^2  ->  1/sqrt(INPUT_DIM) is exactly 1/48
#define OFFSET_TOP   (1.0f / 48.0f)

typedef __attribute__((ext_vector_type(2))) float v2f;
typedef __attribute__((ext_vector_type(8))) float v8f;

// ---------------------------------------------------------------------------
// Kernel A: codes[256] = offset_relu(enc_top @ x), slope 0
// 16 blocks x 128 threads (4 waves). Each block: 16 rows. K split over waves.
// WMMA f32 16x16x4. B carries the x slice replicated into ALL 16 columns, so
// every column of C equals the matvec result -> no lane masking. x staged in
// LDS. Trip count is a compile-time constant so loop control stays scalar.
// ---------------------------------------------------------------------------
__global__ __launch_bounds__(128) void moe_codes_kernel(
    const float* __restrict__ x,
    const float* __restrict__ enc_top,
    float* __restrict__ codes) {
  __shared__ float sX[INPUT_DIM];
  __shared__ float red[4][16];

  const int t       = threadIdx.x;
  const int lane    = t & 31;
  const int wave    = t >> 5;
  const int rowBase = blockIdx.x * 16;
  const int kOfs    = (lane >> 4) * 2;  // lanes 0-15 -> K+0/1, lanes 16-31 -> K+2/3
  const int kStart  = wave * (INPUT_DIM / 4);  // 576 K per wave

  for (int i = t; i < INPUT_DIM; i += 128) sX[i] = x[i];
  __syncthreads();

  const float* aPtr = enc_top + (size_t)(rowBase + (lane & 15)) * INPUT_DIM
                      + kStart + kOfs;
  const float* bPtr = sX + kStart + kOfs;

  v8f c = {};
#pragma unroll 4
  for (int it = 0; it < (INPUT_DIM / 4) / 4; ++it) {  // 144 constant trips
    v2f a, b;
    a.x = aPtr[it * 4];
    a.y = aPtr[it * 4 + 1];
    b.x = bPtr[it * 4];           // replicated across all columns N
    b.y = bPtr[it * 4 + 1];
    c = __builtin_amdgcn_wmma_f32_16x16x4_f32(false, a, false, b,
                                              (short)0, c, false, false);
  }

  // Column N=0 of C: lane 0 holds M=0..7 (c[0..7]), lane 16 holds M=8..15.
  if ((lane & 15) == 0) {
    const int half = lane >> 4;
#pragma unroll
    for (int r = 0; r < 8; ++r) red[wave][half * 8 + r] = c[r];
  }
  __syncthreads();
  if (t < 16) {
    float v = red[0][t] + red[1][t] + red[2][t] + red[3][t];
    codes[rowBase + t] = (v >= OFFSET_TOP) ? v : 0.0f;
  }
}

// ---------------------------------------------------------------------------
// Kernel B: top-4 of 256 (serial; strict '>' matches lax.top_k tie-breaking)
// ---------------------------------------------------------------------------
__global__ void moe_topk_kernel(const float* __restrict__ codes,
                                float* __restrict__ topv,
                                int* __restrict__ topi) {
  if (threadIdx.x == 0) {
    int chosen[TOPK];
#pragma unroll
    for (int j = 0; j < TOPK; ++j) {
      float best = -3.402823e38f;
      int bi = 0;
      for (int i = 0; i < NUM_EXPERTS; ++i) {
        bool skip = false;
        for (int p = 0; p < j; ++p) skip = skip || (chosen[p] == i);
        float v = codes[i];
        if (!skip && v > best) { best = v; bi = i; }
      }
      chosen[j] = bi;
      topv[j] = best;
      topi[j] = bi;
    }
  }
}

// ---------------------------------------------------------------------------
// Kernel C: per selected expert (4 blocks x 256 threads):
//   sub  = W_down[e] @ x                    (FMA, x staged in LDS)
//   ec   = leaky_offset_relu(enc[e] @ sub)  (WMMA f32 16x16x4, replicated B)
//   dec  = enc[e]^T @ ec                    (FMA, coalesced over s)
// ---------------------------------------------------------------------------
__global__ __launch_bounds__(256) void moe_expert_kernel(
    const float* __restrict__ x,
    const float* __restrict__ W_down,
    const float* __restrict__ encw,
    const int* __restrict__ topi,
    float* __restrict__ decOut) {
  __shared__ float sX[INPUT_DIM];
  __shared__ float sSub[SUBSPACE_DIM];
  __shared__ float sCodes[ATOMS];
  __shared__ float sRed[256];

  const int t = threadIdx.x;
  const int e = topi[blockIdx.x];
  const float* W = W_down + (size_t)e * SUBSPACE_DIM * INPUT_DIM;
  const float* E = encw   + (size_t)e * ATOMS * SUBSPACE_DIM;

  for (int i = t; i < INPUT_DIM; i += 256) sX[i] = x[i];
  __syncthreads();

  // ---- step 1: sub[row] = dot(W[row], x); 4 threads per row ----
  {
    const int row  = t >> 2;
    const int part = t & 3;
    const float* Wr = W + (size_t)row * INPUT_DIM + part;
    float acc = 0.0f;
#pragma unroll 4
    for (int i = 0; i < INPUT_DIM / 4; ++i)  // 576 constant trips
      acc += Wr[i * 4] * sX[part + i * 4];
    sRed[t] = acc;
  }
  __syncthreads();
  if ((t & 3) == 0)
    sSub[t >> 2] = sRed[t] + sRed[t + 1] + sRed[t + 2] + sRed[t + 3];
  __syncthreads();

  // ---- step 2: expert codes via WMMA; 8 waves x 4 tiles of 16 rows ----
  const int lane = t & 31;
  const int wave = t >> 5;
  const int kOfs = (lane >> 4) * 2;
#pragma unroll
  for (int i = 0; i < 4; ++i) {
    const int rowBase = (wave * 4 + i) * 16;
    const float* Er = E + (size_t)(rowBase + (lane & 15)) * SUBSPACE_DIM + kOfs;
    v8f c = {};
#pragma unroll
    for (int k0 = 0; k0 < SUBSPACE_DIM; k0 += 4) {
      v2f a, b;
      a.x = Er[k0];
      a.y = Er[k0 + 1];
      b.x = sSub[k0 + kOfs];      // replicated across all columns N
      b.y = sSub[k0 + kOfs + 1];
      c = __builtin_amdgcn_wmma_f32_16x16x4_f32(false, a, false, b,
                                                (short)0, c, false, false);
    }
    if ((lane & 15) == 0) {
      const int half = lane >> 4;
#pragma unroll
      for (int r = 0; r < 8; ++r) {
        float v = c[r];
        v = (v >= OFFSET_TOP) ? v : 0.01f * v;  // leaky offset relu
        sCodes[rowBase + half * 8 + r] = v;
      }
    }
  }
  __syncthreads();

  // ---- step 3: dec[s] = sum_a enc[a][s] * sCodes[a]; 4 chunks of 128 atoms ----
  {
    const int s     = t & 63;
    const int chunk = t >> 6;
    const float* Ec = E + (size_t)(chunk * 128) * SUBSPACE_DIM + s;
    float acc = 0.0f;
#pragma unroll 4
    for (int a = 0; a < 128; ++a)
      acc += Ec[(size_t)a * SUBSPACE_DIM] * sCodes[chunk * 128 + a];
    sRed[t] = acc;
  }
  __syncthreads();
  if (t < SUBSPACE_DIM) {
    float d = sRed[t] + sRed[t + 64] + sRed[t + 128] + sRed[t + 192];
    decOut[blockIdx.x * SUBSPACE_DIM + t] = d;
  }
}

// ---------------------------------------------------------------------------
// Kernel D: out[d] = sum_k ( topv[k]*enc_top[e_k][d] + sum_s W_down[e_k][s][d]*dec[k][s] )
// 9 blocks x 256 threads = 2304 outputs, fully coalesced weight streaming.
// ---------------------------------------------------------------------------
__global__ __launch_bounds__(256) void moe_out_kernel(
    const float* __restrict__ enc_top,
    const float* __restrict__ W_down,
    const float* __restrict__ topv,
    const int* __restrict__ topi,
    const float* __restrict__ dec,
    float* __restrict__ out) {
  __shared__ float sDec[TOPK * SUBSPACE_DIM];
  __shared__ float sV[TOPK];
  __shared__ int   sI[TOPK];

  const int t = threadIdx.x;
  if (t < TOPK * SUBSPACE_DIM) sDec[t] = dec[t];
  if (t < TOPK) { sV[t] = topv[t]; sI[t] = topi[t]; }
  __syncthreads();

  const int d = blockIdx.x * 256 + t;
  float acc = 0.0f;
#pragma unroll
  for (int k = 0; k < TOPK; ++k) {
    const int e = sI[k];
    acc += sV[k] * enc_top[(size_t)e * INPUT_DIM + d];
    const float* Wc = W_down + (size_t)e * SUBSPACE_DIM * INPUT_DIM + d;
#pragma unroll 8
    for (int s = 0; s < SUBSPACE_DIM; ++s)
      acc += Wc[(size_t)s * INPUT_DIM] * sDec[k * SUBSPACE_DIM + s];
  }
  out[d] = acc;
}

// ---------------------------------------------------------------------------
extern "C" void kernel_launch(void* const* d_in, const int* in_sizes, int n_in,
                              void* d_out, int out_size, void* d_ws, size_t ws_size,
                              hipStream_t stream) {
  const float* x        = (const float*)d_in[0];   // [2304]
  const float* enc_top  = (const float*)d_in[1];   // [256, 2304]
  const float* W_down   = (const float*)d_in[2];   // [256, 64, 2304]
  const float* encw     = (const float*)d_in[3];   // [256, 512, 64]
  float* out = (float*)d_out;                      // [2304]

  float* wsf   = (float*)d_ws;
  float* codes = wsf;                  // 256 floats
  float* topv  = wsf + 256;            // 4 floats
  int*   topi  = (int*)(wsf + 260);    // 4 ints
  float* decb  = wsf + 264;            // 4*64 floats

  moe_codes_kernel <<<NUM_EXPERTS / 16, 128, 0, stream>>>(x, enc_top, codes);
  moe_topk_kernel  <<<1, 32, 0, stream>>>(codes, topv, topi);
  moe_expert_kernel<<<TOPK, 256, 0, stream>>>(x, W_down, encw, topi, decb);
  moe_out_kernel   <<<INPUT_DIM / 256, 256, 0, stream>>>(enc_top, W_down, topv, topi, decb, out);
}